// MyFinalNetwork_7258494730827
// MI455X (gfx1250) — compile-verified
//
#include <hip/hip_runtime.h>
#include <hip/hip_bf16.h>

#define NN 50000
#define EE 600000
#define GG 256
#define DN 128
#define DE 32
#define EMB 256

typedef __attribute__((ext_vector_type(2))) float v2f;
typedef __attribute__((ext_vector_type(8))) float v8f;

__device__ __forceinline__ v8f wmma_f32_4(v2f a, v2f b, v8f c) {
    // D = A(16x4, f32) * B(4x16, f32) + C(16x16, f32)
    return __builtin_amdgcn_wmma_f32_16x16x4_f32(false, a, false, b, (short)0, c,
                                                 false, false);
}

// ---------------- feature embedding (one-hot expansion) ----------------
__global__ void build_h_kernel(const float* __restrict__ x, float* __restrict__ h) {
    int gid = blockIdx.x * blockDim.x + threadIdx.x;
    if (gid >= NN * DN) return;
    int i = gid >> 7, j = gid & 127;
    float v;
    if (j < 119) {
        int cat = (int)x[i * 10];
        v = (j == cat) ? 1.0f : 0.0f;
    } else {
        v = x[i * 10 + (j - 118)];
    }
    h[gid] = v;
}

__global__ void build_e_kernel(const float* __restrict__ ea, float* __restrict__ e) {
    int gid = blockIdx.x * blockDim.x + threadIdx.x;
    if (gid >= EE * DE) return;
    int i = gid >> 5, j = gid & 31;
    float v;
    if (j < 22) {
        int cat = (int)ea[i * 11];
        v = (j == cat) ? 1.0f : 0.0f;
    } else {
        v = ea[i * 11 + (j - 21)];
    }
    e[gid] = v;
}

// ---------------- edge transform + gather + relu + scatter-add ----------------
// one wave handles a 16-edge tile x 128 output columns: ep = e@We + be,
// msg = relu(h[src]+ep), agg[dst] += msg (h/agg are L2-resident: 192MB L2)
__global__ void __launch_bounds__(256) edge_msg_kernel(
    const float* __restrict__ e, const float* __restrict__ We,
    const float* __restrict__ be, const float* __restrict__ h,
    const int* __restrict__ src, const int* __restrict__ dst,
    float* __restrict__ agg, int n_tiles)
{
    const int lane = threadIdx.x & 31;
    const int wave = threadIdx.x >> 5;
    const int gw = blockIdx.x * (blockDim.x >> 5) + wave;
    const int nw = gridDim.x * (blockDim.x >> 5);
    const int m = lane & 15;
    const int half = lane >> 4;
    for (int t = gw; t < n_tiles; t += nw) {
        const int base = t * 16;
        v2f a[8];
#pragma unroll
        for (int ks = 0; ks < 8; ++ks)
            a[ks] = *(const v2f*)(e + (base + m) * DE + ks * 4 + half * 2);
        int soff[8], doff[8];
#pragma unroll
        for (int r = 0; r < 8; ++r) {
            int row = base + r + half * 8;
            soff[r] = src[row] * DN;   // precomputed row offsets: keeps address
            doff[r] = dst[row] * DN;   // math out of the per-column-tile loop
        }
#pragma unroll
        for (int ct = 0; ct < 8; ++ct) {
            const int col = ct * 16 + m;
            // issue all 8 gathers as one clause (overlaps with WMMA chain)
            float hs[8];
#pragma unroll
            for (int r = 0; r < 8; ++r) hs[r] = h[soff[r] + col];
            v8f acc = {};
#pragma unroll
            for (int ks = 0; ks < 8; ++ks) {
                const float* wp = We + (ks * 4 + half * 2) * DN + col;
                v2f b;
                b.x = wp[0];
                b.y = wp[DN];
                acc = wmma_f32_4(a[ks], b, acc);
            }
            const float bias = be[col];
#pragma unroll
            for (int r = 0; r < 8; ++r) {
                float msg = acc[r] + bias + hs[r];
                msg = msg > 0.0f ? msg : 0.0f;
                atomicAdd(agg + doff[r] + col, msg);
            }
        }
    }
}

// ---------------- node GEMM z=(h+agg)@Wm+bm, fused BN statistics ----------------
__global__ void __launch_bounds__(256) node_update_kernel(
    const float* __restrict__ h, const float* __restrict__ agg,
    const float* __restrict__ Wm, const float* __restrict__ bm,
    float* __restrict__ z, float* __restrict__ gsum, float* __restrict__ gsq,
    int n_tiles)
{
    __shared__ float sSum[DN];
    __shared__ float sSq[DN];
    if (threadIdx.x < DN) { sSum[threadIdx.x] = 0.0f; sSq[threadIdx.x] = 0.0f; }
    __syncthreads();
    const int lane = threadIdx.x & 31;
    const int wave = threadIdx.x >> 5;
    const int gw = blockIdx.x * (blockDim.x >> 5) + wave;
    const int nw = gridDim.x * (blockDim.x >> 5);
    const int m = lane & 15;
    const int half = lane >> 4;
    float lsum[8], lsq[8];
#pragma unroll
    for (int ct = 0; ct < 8; ++ct) { lsum[ct] = 0.0f; lsq[ct] = 0.0f; }
    for (int t = gw; t < n_tiles; t += nw) {
        const int base = t * 16;
        v2f a[32];
#pragma unroll
        for (int ks = 0; ks < 32; ++ks) {
            int off = (base + m) * DN + ks * 4 + half * 2;
            v2f hv = *(const v2f*)(h + off);
            v2f gv = *(const v2f*)(agg + off);
            a[ks] = hv + gv;
        }
#pragma unroll
        for (int ct = 0; ct < 8; ++ct) {
            const int col = ct * 16 + m;
            v8f acc = {};
#pragma unroll
            for (int ks = 0; ks < 32; ++ks) {
                const float* wp = Wm + (ks * 4 + half * 2) * DN + col;
                v2f b;
                b.x = wp[0];
                b.y = wp[DN];
                acc = wmma_f32_4(a[ks], b, acc);
            }
            const float bias = bm[col];
#pragma unroll
            for (int r = 0; r < 8; ++r) {
                float v = acc[r] + bias;
                z[(base + r + half * 8) * DN + col] = v;
                lsum[ct] += v;
                lsq[ct] += v * v;
            }
        }
    }
#pragma unroll
    for (int ct = 0; ct < 8; ++ct) {
        atomicAdd(&sSum[ct * 16 + m], lsum[ct]);
        atomicAdd(&sSq[ct * 16 + m], lsq[ct]);
    }
    __syncthreads();
    if (threadIdx.x < DN) {
        atomicAdd(&gsum[threadIdx.x], sSum[threadIdx.x]);
        atomicAdd(&gsq[threadIdx.x], sSq[threadIdx.x]);
    }
}

// ---------------- BN coefficient finalize + apply ----------------
__global__ void bn_finalize_kernel(const float* __restrict__ gsum,
                                   const float* __restrict__ gsq,
                                   const float* __restrict__ gamma,
                                   const float* __restrict__ beta,
                                   float* __restrict__ scale,
                                   float* __restrict__ shift, int rows) {
    int c = threadIdx.x;
    float inv = 1.0f / (float)rows;
    float mu = gsum[c] * inv;
    float var = gsq[c] * inv - mu * mu;
    float s = gamma[c] * rsqrtf(var + 1e-5f);
    scale[c] = s;
    shift[c] = beta[c] - mu * s;
}

__global__ void bn_relu_kernel(const float* __restrict__ z,
                               const float* __restrict__ scale,
                               const float* __restrict__ shift,
                               float* __restrict__ hout) {
    int gid = blockIdx.x * blockDim.x + threadIdx.x;
    if (gid >= NN * DN) return;
    int c = gid & 127;
    float v = z[gid] * scale[c] + shift[c];
    hout[gid] = v > 0.0f ? v : 0.0f;
}

// ---------------- global mean pool ----------------
__global__ void pool_acc_kernel(const float* __restrict__ h,
                                const int* __restrict__ batch,
                                float* __restrict__ pooled,
                                float* __restrict__ cnt) {
    int gid = blockIdx.x * blockDim.x + threadIdx.x;
    if (gid >= NN * DN) return;
    int i = gid >> 7, c = gid & 127;
    int g = batch[i];
    atomicAdd(&pooled[g * DN + c], h[gid]);
    if (c == 0) atomicAdd(&cnt[g], 1.0f);
}

__global__ void pool_fin_kernel(float* __restrict__ pooled,
                                const float* __restrict__ cnt) {
    int gid = blockIdx.x * blockDim.x + threadIdx.x;  // GG*DN
    int g = gid >> 7;
    float c = cnt[g];
    c = c > 1.0f ? c : 1.0f;
    pooled[gid] /= c;
}

// ---------------- head: BN stats over rows (single block) ----------------
__global__ void bn_rows_kernel(const float* __restrict__ in, int rows, int cols,
                               const float* __restrict__ gamma,
                               const float* __restrict__ beta,
                               float* __restrict__ scale, float* __restrict__ shift) {
    int c = threadIdx.x;
    if (c >= cols) return;
    float s = 0.0f, q = 0.0f;
    for (int r = 0; r < rows; ++r) {
        float v = in[r * cols + c];
        s += v;
        q += v * v;
    }
    float inv = 1.0f / (float)rows;
    float mu = s * inv;
    float var = q * inv - mu * mu;
    float sc = gamma[c] * rsqrtf(var + 1e-5f);
    scale[c] = sc;
    shift[c] = beta[c] - mu * sc;
}

__global__ void fc1_kernel(const float* __restrict__ pooled,
                           const float* __restrict__ scale0,
                           const float* __restrict__ shift0,
                           const float* __restrict__ W1,
                           const float* __restrict__ b1,
                           float* __restrict__ hh1) {
    __shared__ float row[DN];
    int g = blockIdx.x;
    int j = threadIdx.x;  // 256
    if (j < DN) row[j] = pooled[g * DN + j] * scale0[j] + shift0[j];
    __syncthreads();
    float acc = b1[j];
    for (int c = 0; c < DN; ++c) acc += row[c] * W1[c * EMB + j];
    hh1[g * EMB + j] = acc > 0.0f ? acc : 0.0f;
}

__global__ void fc2_kernel(const float* __restrict__ hh1,
                           const float* __restrict__ scale1,
                           const float* __restrict__ shift1,
                           const float* __restrict__ W2,
                           const float* __restrict__ b2,
                           float* __restrict__ out) {
    int g = threadIdx.x;  // 256
    float acc = b2[0];
    for (int j = 0; j < EMB; ++j)
        acc += (hh1[g * EMB + j] * scale1[j] + shift1[j]) * W2[j];
    out[g] = acc;
}

extern "C" void kernel_launch(void* const* d_in, const int* in_sizes, int n_in,
                              void* d_out, int out_size, void* d_ws, size_t ws_size,
                              hipStream_t stream) {
    const float* x = (const float*)d_in[0];
    const int* edge_index = (const int*)d_in[1];
    const float* edge_attr = (const float*)d_in[2];
    const int* batch = (const int*)d_in[3];
    const int* src = edge_index;
    const int* dst = edge_index + EE;

    auto F = [&](int i) { return (const float*)d_in[i]; };
    const float *We[3], *be[3], *Wm[3], *bm[3], *gvec[3], *bvec[3];
    const float *g0, *b0, *W1, *b1, *g1, *b1n, *W2, *b2;
    if (in_sizes[4] == DN * EMB) {
        // JAX sorted-key pytree flatten: W1,W2,b0,b1,b1n,b2,g0,g1,layers[We,Wm,b,be,bm,g]
        W1 = F(4); W2 = F(5); b0 = F(6); b1 = F(7); b1n = F(8); b2 = F(9);
        g0 = F(10); g1 = F(11);
        for (int l = 0; l < 3; ++l) {
            int o = 12 + 6 * l;
            We[l] = F(o); Wm[l] = F(o + 1); bvec[l] = F(o + 2);
            be[l] = F(o + 3); bm[l] = F(o + 4); gvec[l] = F(o + 5);
        }
    } else {
        // dict insertion order: layers[We,be,Wm,bm,g,b], g0,b0,W1,b1,g1,b1n,W2,b2
        for (int l = 0; l < 3; ++l) {
            int o = 4 + 6 * l;
            We[l] = F(o); be[l] = F(o + 1); Wm[l] = F(o + 2);
            bm[l] = F(o + 3); gvec[l] = F(o + 4); bvec[l] = F(o + 5);
        }
        g0 = F(22); b0 = F(23); W1 = F(24); b1 = F(25);
        g1 = F(26); b1n = F(27); W2 = F(28); b2 = F(29);
    }

    // workspace layout (floats)
    float* ws = (float*)d_ws;
    float* h    = ws;                      // N*128
    float* e    = h + (size_t)NN * DN;     // E*32
    float* agg  = e + (size_t)EE * DE;     // N*128
    float* z    = agg + (size_t)NN * DN;   // N*128
    float* st   = z + (size_t)NN * DN;
    float* gsum   = st;            // 128
    float* gsq    = st + 128;      // 128
    float* scale  = st + 256;      // 128
    float* shift  = st + 384;      // 128
    float* pooled = st + 512;      // 256*128
    float* cnt    = pooled + GG * DN;   // 256
    float* scale0 = cnt + GG;      // 128
    float* shift0 = scale0 + 128;  // 128
    float* hh1    = shift0 + 128;  // 256*256
    float* scale1 = hh1 + GG * EMB;     // 256
    float* shift1 = scale1 + EMB;  // 256

    const int ET = EE / 16;  // 37500 edge tiles
    const int NT = NN / 16;  // 3125 node tiles

    build_h_kernel<<<(NN * DN + 255) / 256, 256, 0, stream>>>(x, h);
    build_e_kernel<<<(EE * DE + 255) / 256, 256, 0, stream>>>(edge_attr, e);

    for (int l = 0; l < 3; ++l) {
        hipMemsetAsync(agg, 0, (size_t)NN * DN * sizeof(float), stream);
        hipMemsetAsync(gsum, 0, 256 * sizeof(float), stream);
        edge_msg_kernel<<<(ET + 7) / 8, 256, 0, stream>>>(e, We[l], be[l], h, src,
                                                          dst, agg, ET);
        node_update_kernel<<<(NT + 7) / 8, 256, 0, stream>>>(h, agg, Wm[l], bm[l],
                                                             z, gsum, gsq, NT);
        bn_finalize_kernel<<<1, DN, 0, stream>>>(gsum, gsq, gvec[l], bvec[l],
                                                 scale, shift, NN);
        bn_relu_kernel<<<(NN * DN + 255) / 256, 256, 0, stream>>>(z, scale, shift, h);
    }

    hipMemsetAsync(pooled, 0, (size_t)(GG * DN + GG) * sizeof(float), stream);
    pool_acc_kernel<<<(NN * DN + 255) / 256, 256, 0, stream>>>(h, batch, pooled, cnt);
    pool_fin_kernel<<<(GG * DN) / 256, 256, 0, stream>>>(pooled, cnt);
    bn_rows_kernel<<<1, DN, 0, stream>>>(pooled, GG, DN, g0, b0, scale0, shift0);
    fc1_kernel<<<GG, EMB, 0, stream>>>(pooled, scale0, shift0, W1, b1, hh1);
    bn_rows_kernel<<<1, EMB, 0, stream>>>(hh1, GG, EMB, g1, b1n, scale1, shift1);
    fc2_kernel<<<1, GG, 0, stream>>>(hh1, scale1, shift1, W2, b2, (float*)d_out);
}